// L2ClusterCentroid_90924457656744
// MI455X (gfx1250) — compile-verified
//
#include <hip/hip_runtime.h>
#include <math.h>

typedef __attribute__((ext_vector_type(2))) float v2f;
typedef __attribute__((ext_vector_type(8))) float v8f;

#define DDIM 128
#define NCLS 100
#define CPAD 112            // 7 WMMA tiles of 16 classes
#define NTILES 7

// d_ws float layout: [0, CPAD*DDIM) class sums ; [CPAD*DDIM, CPAD*DDIM+CPAD) counts
#define WS_SUMS 0
#define WS_CNT  (CPAD * DDIM)
#define WS_TOTAL (CPAD * DDIM + CPAD)

__global__ void zero_ws_kernel(float* __restrict__ ws) {
  int i = blockIdx.x * blockDim.x + threadIdx.x;
  if (i < WS_TOTAL) ws[i] = 0.0f;
}

// Streaming pass: wave-per-row argmax + LDS segment accumulation.
// 912 MB of logits+embedding streamed exactly once -> HBM-bound (~39 us at 23.3 TB/s).
__global__ __launch_bounds__(512) void accum_kernel(
    const float* __restrict__ emb, const float* __restrict__ logits,
    float* __restrict__ gsums, float* __restrict__ gcnt, int nrows) {
  __shared__ float s_sums[NCLS * DDIM];   // 51.2 KB of the 320 KB WGP LDS
  __shared__ float s_cnt[NCLS];

  for (int i = threadIdx.x; i < NCLS * DDIM; i += blockDim.x) s_sums[i] = 0.0f;
  for (int i = threadIdx.x; i < NCLS; i += blockDim.x) s_cnt[i] = 0.0f;
  __syncthreads();

  const int lane = threadIdx.x & 31;          // wave32
  const int wid  = threadIdx.x >> 5;
  const int wpb  = blockDim.x >> 5;
  long gw = (long)blockIdx.x * wpb + wid;
  long nw = (long)gridDim.x * wpb;

  for (long row = gw; row < (long)nrows; row += nw) {
    // ---- argmax over 100 logits (first-max-index tie-break, matches jnp.argmax)
    const float* lg = logits + row * (long)NCLS;
    float v = -__builtin_inff();
    int bi = 0x7fffffff;
    for (int j = lane; j < NCLS; j += 32) {
      float x = lg[j];
      if (x > v) { v = x; bi = j; }
    }
#pragma unroll
    for (int off = 16; off > 0; off >>= 1) {
      float ov = __shfl_xor(v, off, 32);
      int   oi = __shfl_xor(bi, off, 32);
      if (ov > v || (ov == v && oi < bi)) { v = ov; bi = oi; }
    }
    // ---- accumulate embedding row into LDS; lane L -> dwords L+32k:
    // bank (L+32k) mod 64 is unique per lane => conflict-free ds_add_f32.
    const float* er = emb + row * (long)DDIM;
    float* dst = s_sums + bi * DDIM;
#pragma unroll
    for (int k = 0; k < DDIM / 32; ++k) {
      atomicAdd(&dst[lane + 32 * k], er[lane + 32 * k]);
    }
    if (lane == 0) atomicAdd(&s_cnt[bi], 1.0f);
  }

  __syncthreads();
  // ---- per-block flush into global accumulator (~12.9K f32 atomics / block,
  // 12.8K distinct addresses spread across L2 slices; ~6.6M total, well under
  // the HBM-time budget).
  for (int i = threadIdx.x; i < NCLS * DDIM; i += blockDim.x) {
    float x = s_sums[i];
    if (x != 0.0f) atomicAdd(&gsums[i], x);
  }
  for (int i = threadIdx.x; i < NCLS; i += blockDim.x) {
    float x = s_cnt[i];
    if (x != 0.0f) atomicAdd(&gcnt[i], x);
  }
}

// Finalize: per 16-class tile, sq distances = diag(Delta @ Delta^T) via
// V_WMMA_F32_16X16X4_F32 accumulated over K=128. 32-bit A(16x4) and B(4x16)
// register layouts coincide (VGPR0: K=0 lanes0-15 / K=2 lanes16-31; VGPR1:
// K=1 / K=3), so one register pair feeds both operands. Out-of-range classes
// (100..111 in tile 6) are handled with a multiplicative mask so EXEC stays
// all-1s through the whole WMMA chain (no saveexec per step).
__global__ __launch_bounds__(256) void finalize_wmma_kernel(
    const float* __restrict__ centers, const float* __restrict__ gsums,
    const float* __restrict__ gcnt, float* __restrict__ out) {
  const int lane = threadIdx.x & 31;
  const int wid  = threadIdx.x >> 5;
  if (wid >= NTILES) return;                 // whole-wave exit

  const int hi   = lane >> 4;                // half-wave select (K offset 0 or 2)
  const int m    = lane & 15;                // matrix row M within tile
  const int cls  = wid * 16 + m;
  const int clsc = (cls < NCLS) ? cls : 0;   // clamped for always-in-bounds loads
  const float vm = (cls < NCLS) ? 1.0f : 0.0f;
  const float inv = 1.0f / fmaxf(gcnt[clsc], 1.0f);

  const float* crow = centers + (long)clsc * DDIM;
  const float* srow = gsums   + (long)clsc * DDIM;

  v8f acc = {};
#pragma unroll
  for (int kk = 0; kk < DDIM; kk += 4) {
    const int k0 = kk + 2 * hi;
    float d0 = (crow[k0]     - srow[k0]     * inv) * vm;
    float d1 = (crow[k0 + 1] - srow[k0 + 1] * inv) * vm;
    v2f a; a.x = d0; a.y = d1;
    acc = __builtin_amdgcn_wmma_f32_16x16x4_f32(
        /*neg_a=*/false, a, /*neg_b=*/false, a,
        /*c_mod=*/(short)0, acc, /*reuse_a=*/false, /*reuse_b=*/false);
  }

  // Diagonal of 16x16 f32 C/D: VGPR r holds (M=r, N=lane) lanes0-15 and
  // (M=r+8, N=lane-16) lanes16-31 -> diag at lane==r and lane==r+24.
  int r = -1, oc = -1;
  if (lane < 8)        { r = lane;      oc = wid * 16 + lane; }
  else if (lane >= 24) { r = lane - 24; oc = wid * 16 + (lane - 16); }
  if (r >= 0 && oc < NCLS) {
    float sq = 0.0f;
#pragma unroll
    for (int j = 0; j < 8; ++j) if (j == r) sq = acc[j];
    float cnt  = gcnt[oc];
    float dist = (sq > 0.0f) ? sqrtf(sq) : 0.0f;
    out[oc] = (cnt > 0.0f) ? dist : 0.0f;
  }
}

extern "C" void kernel_launch(void* const* d_in, const int* in_sizes, int n_in,
                              void* d_out, int out_size, void* d_ws, size_t ws_size,
                              hipStream_t stream) {
  const float* emb     = (const float*)d_in[0];  // [N, 128]
  const float* centers = (const float*)d_in[1];  // [100, 128]
  const float* logits  = (const float*)d_in[2];  // [N, 100]
  float* out = (float*)d_out;                    // [100]
  float* ws  = (float*)d_ws;

  const int nrows = in_sizes[0] / DDIM;
  float* gsums = ws + WS_SUMS;
  float* gcnt  = ws + WS_CNT;

  zero_ws_kernel<<<(WS_TOTAL + 255) / 256, 256, 0, stream>>>(ws);
  accum_kernel<<<512, 512, 0, stream>>>(emb, logits, gsums, gcnt, nrows);
  finalize_wmma_kernel<<<1, 256, 0, stream>>>(centers, gsums, gcnt, out);
}